// GNNClassifier_88648124990764
// MI455X (gfx1250) — compile-verified
//
#include <hip/hip_runtime.h>
#include <hip/hip_bf16.h>

#define N_NODES   100000
#define N_EDGES   1600000
#define HID       128
#define POS_CAP   512
#define N_CLASSES 10
#define N_GRAPHS  512
#define BN_EPS    1e-5f
#define MT        5            // M-tiles per wave: 80 rows; 100000 = 1250*80 exactly

typedef float v2f __attribute__((ext_vector_type(2)));
typedef float v8f __attribute__((ext_vector_type(8)));

// Hardware f32 atomic add (global_atomic_add_f32); all hot targets are L2-resident.
__device__ __forceinline__ void atomAddF(float* p, float v) {
    unsafeAtomicAdd(p, v);
}

// ---------------------------------------------------------------------------
// 1) Embedding gather: x[n] = shape_tab[sid] + col_tab[cid] + pos_tab[min(pid,512)]
// ---------------------------------------------------------------------------
__global__ __launch_bounds__(256)
void embed_kernel(const int* __restrict__ sid, const int* __restrict__ cid,
                  const int* __restrict__ pid,
                  const float* __restrict__ stab, const float* __restrict__ ctab,
                  const float* __restrict__ ptab, float* __restrict__ x)
{
    int idx = blockIdx.x * blockDim.x + threadIdx.x;
    int n = idx >> 5, q = idx & 31;
    if (n >= N_NODES) return;
    int s = sid[n];
    int c = cid[n];
    int p = min(pid[n], POS_CAP);
    const float4 a = *(const float4*)(stab + (size_t)s * HID + q * 4);
    const float4 b = *(const float4*)(ctab + (size_t)c * HID + q * 4);
    const float4 d = *(const float4*)(ptab + (size_t)p * HID + q * 4);
    float4 o;
    o.x = a.x + b.x + d.x;  o.y = a.y + b.y + d.y;
    o.z = a.z + b.z + d.z;  o.w = a.w + b.w + d.w;
    *(float4*)(x + (size_t)n * HID + q * 4) = o;
}

// ---------------------------------------------------------------------------
// 2) Edge scatter: one wave per edge. 32 lanes x float4 cover the 128-float row.
// ---------------------------------------------------------------------------
__global__ __launch_bounds__(256)
void scatter_sum_kernel(const int* __restrict__ ei, const float* __restrict__ x,
                        float* __restrict__ sums, float* __restrict__ cnt,
                        int countDegrees)
{
    int e = blockIdx.x * 8 + (threadIdx.x >> 5);
    if (e >= N_EDGES) return;
    int lane = threadIdx.x & 31;
    int src = ei[e];             // edge_index row 0
    int dst = ei[N_EDGES + e];   // edge_index row 1
    const float4 v = *(const float4*)(x + (size_t)src * HID + lane * 4);
    float* o = sums + (size_t)dst * HID + lane * 4;
    atomAddF(o + 0, v.x);
    atomAddF(o + 1, v.y);
    atomAddF(o + 2, v.z);
    atomAddF(o + 3, v.w);
    if (countDegrees && lane == 0) atomAddF(cnt + dst, 1.0f);
}

// ---------------------------------------------------------------------------
// 3) mean = sums / max(cnt, 1)   (in place)
// ---------------------------------------------------------------------------
__global__ __launch_bounds__(256)
void mean_div_kernel(float* __restrict__ sums, const float* __restrict__ cnt)
{
    int idx = blockIdx.x * blockDim.x + threadIdx.x;
    int n = idx >> 5, q = idx & 31;
    if (n >= N_NODES) return;
    float inv = 1.0f / fmaxf(cnt[n], 1.0f);
    float4* p = (float4*)(sums + (size_t)n * HID) + q;
    float4 v = *p;
    v.x *= inv; v.y *= inv; v.z *= inv; v.w *= inv;
    *p = v;
}

// ---------------------------------------------------------------------------
// 4) SAGE linear via fp32 WMMA, 80x16 super-tile per wave:
//    out[m][n] = sum_k agg[m][k]*wl[n][k] + sum_k xr[m][k]*wr[n][k] + bias[n]
//    - 5 independent accumulators -> WMMA dependency distance 5 (pipelined XDL)
//    - weight fragments loaded once per K-step, reused by all 5 M-tiles
//    - tile offsets are compile-time constants (t*8192 B) folded into IOFFSET
//    - EXEC stays all-1s: 100000 = 1250*80, 128 = 8*16, no predication anywhere
//    Fragment layouts (32-bit, wave32):
//      A (16x4):  lanes 0-15 -> K=k0+{0,1}; lanes 16-31 -> K=k0+{2,3}; M=lane&15
//      B (4x16):  same K split; N=lane&15
//      C/D (16x16, 8 VGPRs): element r -> row r + 8*(lane>=16), col lane&15
// ---------------------------------------------------------------------------
__global__ __launch_bounds__(256)
void sage_linear_kernel(const float* __restrict__ agg, const float* __restrict__ xr,
                        const float* __restrict__ wl,  const float* __restrict__ wr,
                        const float* __restrict__ bias, float* __restrict__ out)
{
    const int lane  = threadIdx.x & 31;
    const int wave  = threadIdx.x >> 5;        // 8 waves -> 8 N-tiles (128 = 8*16)
    const int m0    = blockIdx.x * (16 * MT);  // 1250 blocks cover all nodes
    const int n0    = wave * 16;
    const int l16   = lane & 15;
    const int khalf = (lane >> 4) * 2;         // 0 or 2

    const float* wlp   = wl  + (size_t)(n0 + l16) * HID + khalf;
    const float* wrp   = wr  + (size_t)(n0 + l16) * HID + khalf;
    const float* abase = agg + (size_t)(m0 + l16) * HID + khalf;
    const float* xbase = xr  + (size_t)(m0 + l16) * HID + khalf;

    v8f acc[MT];
#pragma unroll
    for (int t = 0; t < MT; ++t) acc[t] = (v8f){};

#pragma unroll 2
    for (int k0 = 0; k0 < HID; k0 += 4) {
        // B fragments: loaded once, reused by all MT tiles
        v2f bl = *(const v2f*)(wlp + k0);
        v2f br = *(const v2f*)(wrp + k0);
        // A fragments for all tiles (constant byte offsets t*16*HID*4 = t*8192)
        v2f a[MT], xv[MT];
#pragma unroll
        for (int t = 0; t < MT; ++t) a[t]  = *(const v2f*)(abase + k0 + t * (16 * HID));
#pragma unroll
        for (int t = 0; t < MT; ++t) xv[t] = *(const v2f*)(xbase + k0 + t * (16 * HID));
        // 10 WMMAs rotating over 5 accumulators -> independent back-to-back issue
#pragma unroll
        for (int t = 0; t < MT; ++t)
            acc[t] = __builtin_amdgcn_wmma_f32_16x16x4_f32(false, a[t], false, bl,
                                                           (short)0, acc[t], false, false);
#pragma unroll
        for (int t = 0; t < MT; ++t)
            acc[t] = __builtin_amdgcn_wmma_f32_16x16x4_f32(false, xv[t], false, br,
                                                           (short)0, acc[t], false, false);
    }

    const float bn    = bias[n0 + l16];
    const int   col   = n0 + l16;
    float* obase = out + (size_t)(m0 + (lane >> 4) * 8) * HID + col;
#pragma unroll
    for (int t = 0; t < MT; ++t) {
#pragma unroll
        for (int r = 0; r < 8; ++r)
            obase[(size_t)(t * 16 + r) * HID] = acc[t][r] + bn;
    }
}

// ---------------------------------------------------------------------------
// 5) BatchNorm stats: per-feature sum and sum-of-squares over all nodes.
// ---------------------------------------------------------------------------
__global__ __launch_bounds__(256)
void bn_stats_kernel(const float* __restrict__ h, float* __restrict__ stats)
{
    __shared__ float ssum[256];
    __shared__ float ssq[256];
    const int f    = threadIdx.x & 127;
    const int half = threadIdx.x >> 7;
    float s = 0.0f, ss = 0.0f;
    for (int n = blockIdx.x * 2 + half; n < N_NODES; n += gridDim.x * 2) {
        float v = h[(size_t)n * HID + f];
        s += v; ss += v * v;
    }
    ssum[threadIdx.x] = s; ssq[threadIdx.x] = ss;
    __syncthreads();
    if (threadIdx.x < 128) {
        atomAddF(stats + f,        ssum[threadIdx.x] + ssum[threadIdx.x + 128]);
        atomAddF(stats + 128 + f,  ssq[threadIdx.x]  + ssq[threadIdx.x + 128]);
    }
}

// ---------------------------------------------------------------------------
// 6) Fused BN (training-mode) + ReLU, in place.
// ---------------------------------------------------------------------------
__global__ __launch_bounds__(256)
void bn_relu_kernel(float* __restrict__ h, const float* __restrict__ stats,
                    const float* __restrict__ gamma, const float* __restrict__ beta)
{
    int idx = blockIdx.x * blockDim.x + threadIdx.x;
    int n = idx >> 5, q = idx & 31;
    if (n >= N_NODES) return;
    const float invN = 1.0f / (float)N_NODES;
    float4* p = (float4*)(h + (size_t)n * HID) + q;
    float4 v = *p;
    float r[4] = {v.x, v.y, v.z, v.w};
#pragma unroll
    for (int j = 0; j < 4; ++j) {
        int f = q * 4 + j;
        float mu  = stats[f] * invN;
        float var = stats[128 + f] * invN - mu * mu;
        float sc  = gamma[f] * __frsqrt_rn(var + BN_EPS);
        float sh  = beta[f] - mu * sc;
        r[j] = fmaxf(r[j] * sc + sh, 0.0f);
    }
    v.x = r[0]; v.y = r[1]; v.z = r[2]; v.w = r[3];
    *p = v;
}

// ---------------------------------------------------------------------------
// 7) Graph pooling: pooled[batch[n]] += h[n]  (256 KB target, L2-resident)
// ---------------------------------------------------------------------------
__global__ __launch_bounds__(256)
void pool_kernel(const float* __restrict__ h, const int* __restrict__ batch,
                 float* __restrict__ pooled)
{
    int idx = blockIdx.x * blockDim.x + threadIdx.x;
    int n = idx >> 5, q = idx & 31;
    if (n >= N_NODES) return;
    int g = batch[n];
    const float4 v = *(const float4*)(h + (size_t)n * HID + q * 4);
    float* o = pooled + (size_t)g * HID + q * 4;
    atomAddF(o + 0, v.x);
    atomAddF(o + 1, v.y);
    atomAddF(o + 2, v.z);
    atomAddF(o + 3, v.w);
}

// ---------------------------------------------------------------------------
// 8) Output linear: out[g][c] = pooled[g] . wout[c] + bout[c]   (512x10, tiny)
// ---------------------------------------------------------------------------
__global__ __launch_bounds__(256)
void out_linear_kernel(const float* __restrict__ pooled, const float* __restrict__ wout,
                       const float* __restrict__ bout, float* __restrict__ out)
{
    int idx = blockIdx.x * blockDim.x + threadIdx.x;
    if (idx >= N_GRAPHS * N_CLASSES) return;
    int g = idx / N_CLASSES, c = idx % N_CLASSES;
    const float* p = pooled + (size_t)g * HID;
    const float* w = wout + (size_t)c * HID;
    float acc = bout[c];
#pragma unroll 8
    for (int k = 0; k < HID; ++k) acc = fmaf(p[k], w[k], acc);
    out[idx] = acc;
}

// ---------------------------------------------------------------------------
// Host-side orchestration (all on `stream`; graph-capture safe)
// ---------------------------------------------------------------------------
extern "C" void kernel_launch(void* const* d_in, const int* in_sizes, int n_in,
                              void* d_out, int out_size, void* d_ws, size_t ws_size,
                              hipStream_t stream)
{
    const int*   sid   = (const int*)  d_in[0];
    const int*   cid   = (const int*)  d_in[1];
    const int*   pid   = (const int*)  d_in[2];
    const int*   ei    = (const int*)  d_in[3];
    const int*   batch = (const int*)  d_in[4];
    const float* stab  = (const float*)d_in[5];
    const float* ctab  = (const float*)d_in[6];
    const float* ptab  = (const float*)d_in[7];
    const float* w1l   = (const float*)d_in[8];
    const float* b1l   = (const float*)d_in[9];
    const float* w1r   = (const float*)d_in[10];
    const float* g1    = (const float*)d_in[11];
    const float* be1   = (const float*)d_in[12];
    const float* w2l   = (const float*)d_in[13];
    const float* b2l   = (const float*)d_in[14];
    const float* w2r   = (const float*)d_in[15];
    const float* g2    = (const float*)d_in[16];
    const float* be2   = (const float*)d_in[17];
    const float* wout  = (const float*)d_in[18];
    const float* bout  = (const float*)d_in[19];
    float* out = (float*)d_out;

    // Workspace layout (floats). X reused as layer-2 output after layer-1 GEMM.
    const size_t NF = (size_t)N_NODES * HID;        // 12.8M floats
    float* X      = (float*)d_ws;                   // embeddings, then layer-2 out
    float* AGG    = X + NF;                         // scatter sums -> mean
    float* H      = AGG + NF;                       // layer-1 out (post BN-ReLU)
    float* CNT    = H + NF;                         // [N_NODES] degrees
    float* STATS  = CNT + N_NODES;                  // [256]
    float* POOLED = STATS + 256;                    // [N_GRAPHS*HID]

    const int nodeQThreads = N_NODES * 32;
    const int nodeQBlocks  = (nodeQThreads + 255) / 256;   // 12500
    const int edgeBlocks   = (N_EDGES + 7) / 8;            // 200000
    const int gemmBlocks   = N_NODES / (16 * MT);          // 1250 (exact)

    // ---- init scratch ----
    hipMemsetAsync(AGG,   0, NF * sizeof(float), stream);
    hipMemsetAsync(CNT,   0, N_NODES * sizeof(float), stream);
    hipMemsetAsync(STATS, 0, 256 * sizeof(float), stream);
    hipMemsetAsync(POOLED,0, (size_t)N_GRAPHS * HID * sizeof(float), stream);

    // ---- embeddings ----
    embed_kernel<<<nodeQBlocks, 256, 0, stream>>>(sid, cid, pid, stab, ctab, ptab, X);

    // ---- layer 1 ----
    scatter_sum_kernel<<<edgeBlocks, 256, 0, stream>>>(ei, X, AGG, CNT, 1);
    mean_div_kernel<<<nodeQBlocks, 256, 0, stream>>>(AGG, CNT);
    sage_linear_kernel<<<gemmBlocks, 256, 0, stream>>>(AGG, X, w1l, w1r, b1l, H);
    bn_stats_kernel<<<400, 256, 0, stream>>>(H, STATS);
    bn_relu_kernel<<<nodeQBlocks, 256, 0, stream>>>(H, STATS, g1, be1);

    // ---- layer 2 (degrees unchanged; reuse CNT) ----
    hipMemsetAsync(AGG,   0, NF * sizeof(float), stream);
    hipMemsetAsync(STATS, 0, 256 * sizeof(float), stream);
    scatter_sum_kernel<<<edgeBlocks, 256, 0, stream>>>(ei, H, AGG, CNT, 0);
    mean_div_kernel<<<nodeQBlocks, 256, 0, stream>>>(AGG, CNT);
    sage_linear_kernel<<<gemmBlocks, 256, 0, stream>>>(AGG, H, w2l, w2r, b2l, X);
    bn_stats_kernel<<<400, 256, 0, stream>>>(X, STATS);
    bn_relu_kernel<<<nodeQBlocks, 256, 0, stream>>>(X, STATS, g2, be2);

    // ---- pooling + classifier ----
    pool_kernel<<<nodeQBlocks, 256, 0, stream>>>(X, batch, POOLED);
    out_linear_kernel<<<(N_GRAPHS * N_CLASSES + 255) / 256, 256, 0, stream>>>(
        POOLED, wout, bout, out);
}